// DirectDetectionLoss_16973710754131
// MI455X (gfx1250) — compile-verified
//
#include <hip/hip_runtime.h>
#include <math.h>

// Problem constants (match reference)
#define Bc 8
#define Cc 48
#define Hc 128
#define Wc 128
#define Kc 32
#define HWc (Hc * Wc)                 // 16384
#define NCELLS (Bc * Cc * Hc * Wc)    // 6,291,456
#define NPAIR (Bc * Kc)               // 256
#define TPB 256
#define NB_LOSS (NCELLS / TPB)        // 24576
#define WAVES_PER_BLK (TPB / 32)      // 8 (wave32)
#define ELEMS_PER_WAVE (HWc / WAVES_PER_BLK)   // 2048
#define WMMA_STEPS (ELEMS_PER_WAVE / 64)       // 32

typedef __attribute__((address_space(3))) char lds_char;
typedef __attribute__((ext_vector_type(2))) float v2f;
typedef __attribute__((ext_vector_type(8))) float v8f;

// ---------------------------------------------------------------------------
// gfx1250 async-tensor path: copy 16B (one box) from global into LDS via the
// ASYNCcnt-tracked async load unit, then wait the counter to 0.
// ---------------------------------------------------------------------------
__device__ __forceinline__ void async_box_to_lds(void* lds_dst, const void* gsrc) {
    unsigned lds_off = (unsigned)(unsigned long long)(lds_char*)lds_dst;
    unsigned long long ga = (unsigned long long)gsrc;
    asm volatile("global_load_async_to_lds_b128 %0, %1, off"
                 :: "v"(lds_off), "v"(ga)
                 : "memory");
    asm volatile("s_wait_asynccnt 0" ::: "memory");
}

// GIoU exactly mirroring the reference expression order
__device__ __forceinline__ float giou_f(float4 a, float4 b) {
    float area_a = (a.z - a.x) * (a.w - a.y);
    float area_b = (b.z - b.x) * (b.w - b.y);
    float iw = fmaxf(fminf(a.z, b.z) - fmaxf(a.x, b.x), 0.0f);
    float ih = fmaxf(fminf(a.w, b.w) - fmaxf(a.y, b.y), 0.0f);
    float inter = iw * ih;
    float uni = area_a + area_b - inter;
    float iou = inter / uni;
    float ew = fmaxf(a.z, b.z) - fminf(a.x, b.x);
    float eh = fmaxf(a.w, b.w) - fminf(a.y, b.y);
    float enc = ew * eh;
    return iou - (enc - uni) / enc;
}

// Deterministic block reductions (fixed tree, 256 threads)
__device__ __forceinline__ float blk_sum_f(float v, float* sh) {
    const int t = threadIdx.x;
    sh[t] = v;
    __syncthreads();
    for (int s = TPB / 2; s >= 1; s >>= 1) {
        if (t < s) sh[t] += sh[t + s];
        __syncthreads();
    }
    float r = sh[0];
    __syncthreads();
    return r;
}

__device__ __forceinline__ int blk_sum_i(int v, int* sh) {
    const int t = threadIdx.x;
    sh[t] = v;
    __syncthreads();
    for (int s = TPB / 2; s >= 1; s >>= 1) {
        if (t < s) sh[t] += sh[t + s];
        __syncthreads();
    }
    int r = sh[0];
    __syncthreads();
    return r;
}

// ---------------------------------------------------------------------------
// K1: matched map <- -1
// ---------------------------------------------------------------------------
__global__ void k_init(int* __restrict__ matched) {
    int i = blockIdx.x * TPB + threadIdx.x;
    if (i < NCELLS) matched[i] = -1;
}

// ---------------------------------------------------------------------------
// K2: per-(b,k) GIoU max + first-argmax over the (b, label) plane
// ---------------------------------------------------------------------------
__global__ void k_match(const float* __restrict__ pred,
                        const float* __restrict__ gtb,
                        const int* __restrict__ glab,
                        float* __restrict__ gmax,
                        int* __restrict__ amax) {
    __shared__ float sv[TPB];
    __shared__ int   si[TPB];
    __shared__ float4 boxsh;
    const int pair = blockIdx.x;          // 0..255
    const int b = pair / Kc;
    const int c = glab[pair];
    if (threadIdx.x == 0)
        async_box_to_lds(&boxsh, (const void*)(((const float4*)gtb) + pair));
    __syncthreads();
    const float4 bt = boxsh;
    const float4* plane = ((const float4*)pred) + (size_t)(b * Cc + c) * HWc;

    float best = -INFINITY;
    int bidx = 0;
    for (int cell = threadIdx.x; cell < HWc; cell += TPB) {
        if (cell + 4 * TPB < HWc)
            __builtin_prefetch(plane + cell + 4 * TPB, 0, 0);   // global_prefetch_b8
        float g = giou_f(plane[cell], bt);
        if (g > best) { best = g; bidx = cell; }                // keeps first max per thread
    }
    const int t = threadIdx.x;
    sv[t] = best; si[t] = bidx;
    __syncthreads();
    for (int s = TPB / 2; s >= 1; s >>= 1) {
        if (t < s) {
            float ov = sv[t + s]; int oi = si[t + s];
            if (ov > sv[t] || (ov == sv[t] && oi < si[t])) { sv[t] = ov; si[t] = oi; }
        }
        __syncthreads();
    }
    if (t == 0) { gmax[pair] = sv[0]; amax[pair] = si[0]; }
}

// ---------------------------------------------------------------------------
// K3: scatter-max of k into the <=4x4 trunc window (one thread per pair)
// ---------------------------------------------------------------------------
__global__ void k_scatter(const float* __restrict__ gmax,
                          const int* __restrict__ amax,
                          const int* __restrict__ glab,
                          int* __restrict__ matched) {
    const int pair = threadIdx.x;         // 256 threads, 1 block
    const int b = pair / Kc;
    const int k = pair % Kc;
    if (gmax[pair] > 0.3f) {
        const int a = amax[pair];
        float mi = (float)(a / Wc);
        float mj = (float)(a % Wc);
        float i_lo = truncf(mi - 1.5f);
        float i_hi = fminf((float)(Hc - 1), truncf(mi + 1.5f));
        float j_lo = truncf(mj - 1.5f);
        float j_hi = fminf((float)(Wc - 1), truncf(mj + 1.5f));
        int ilo = (int)fmaxf(0.0f, i_lo), ihi = (int)i_hi;
        int jlo = (int)fmaxf(0.0f, j_lo), jhi = (int)j_hi;
        const int c = glab[pair];
        for (int i = ilo; i <= ihi; ++i)
            for (int j = jlo; j <= jhi; ++j)
                atomicMax(&matched[((b * Cc + c) * Hc + i) * Wc + j], k);
    }
}

// ---------------------------------------------------------------------------
// K4: per-cell L1/GIoU (positives only) + focal term (all cells), block partials
// ---------------------------------------------------------------------------
__global__ void k_loss(const float4* __restrict__ pred,
                       const float* __restrict__ conf,
                       const float4* __restrict__ gtb,
                       const int* __restrict__ matched,
                       float* __restrict__ p_l1,
                       float* __restrict__ p_gi,
                       float* __restrict__ p_cf,
                       int* __restrict__ p_np) {
    __shared__ float shf[TPB];
    __shared__ int shi[TPB];
    const int t = threadIdx.x;
    const int n = blockIdx.x * TPB + t;
    const int m = matched[n];
    const float cf = conf[n];

    float l1 = 0.0f, gl = 0.0f, tt = 0.0f;
    int np = 0;
    if (m >= 0) {
        const int b = n / (Cc * HWc);
        float4 pb = pred[n];
        float4 gt = gtb[b * Kc + m];
        l1 = 0.25f * (fabsf(pb.x - gt.x) + fabsf(pb.y - gt.y) +
                      fabsf(pb.z - gt.z) + fabsf(pb.w - gt.w));
        gl = 1.0f - giou_f(pb, gt);
        tt = 1.0f;
        np = 1;
    }
    // focal loss term
    float p  = fminf(fmaxf(cf, 1e-6f), 1.0f - 1e-6f);
    float ce = -(tt * logf(p) + (1.0f - tt) * log1pf(-p));
    float p_t = p * tt + (1.0f - p) * (1.0f - tt);
    float at  = 0.25f * tt + 0.75f * (1.0f - tt);
    float om  = 1.0f - p_t;
    float fl  = at * om * om * ce;

    float s_l1 = blk_sum_f(l1, shf);
    float s_gi = blk_sum_f(gl, shf);
    float s_cf = blk_sum_f(fl, shf);
    int   s_np = blk_sum_i(np, shi);
    if (t == 0) {
        p_l1[blockIdx.x] = s_l1;
        p_gi[blockIdx.x] = s_gi;
        p_cf[blockIdx.x] = s_cf;
        p_np[blockIdx.x] = s_np;
    }
}

// ---------------------------------------------------------------------------
// K5: CAM term per (b,k) via the FP32 tensor pipe.
// Trick: with B == all-ones, sum(D) == 16 * sum(A elements) independent of the
// WMMA VGPR<->matrix mapping, so v_wmma_f32_16x16x4_f32 acts as an exact f32
// reduction engine. Two accumulator chains: raw (plane total) and masked
// (rectangle sum, mask applied per element before the matrix op).
// ---------------------------------------------------------------------------
__global__ void k_cam(const float* __restrict__ cam,
                      const float* __restrict__ gtb,
                      const int* __restrict__ glab,
                      float* __restrict__ terms) {
    __shared__ float wtot[WAVES_PER_BLK];
    __shared__ float wrect[WAVES_PER_BLK];
    __shared__ float4 boxsh;
    const int pair = blockIdx.x;
    const int b = pair / Kc;
    const int c = glab[pair];
    if (threadIdx.x == 0)
        async_box_to_lds(&boxsh, (const void*)(((const float4*)gtb) + pair));
    __syncthreads();
    const float4 g = boxsh;   // (xmin, ymin, xmax, ymax)

    const int ilo = (int)fmaxf(0.0f, truncf(g.y * (float)Hc));
    const int ihi = (int)fminf((float)(Hc - 1), truncf(g.w * (float)Hc));
    const int jlo = (int)fmaxf(0.0f, truncf(g.x * (float)Wc));
    const int jhi = (int)fminf((float)(Wc - 1), truncf(g.z * (float)Wc));

    const float* plane = cam + (size_t)(b * Cc + c) * HWc;
    const int lane = threadIdx.x & 31;
    const int wave = threadIdx.x >> 5;
    const int wbase = wave * ELEMS_PER_WAVE;

    const v2f bones = {1.0f, 1.0f};       // all-ones B matrix (layout-agnostic)
    v8f c_tot  = {0.f, 0.f, 0.f, 0.f, 0.f, 0.f, 0.f, 0.f};
    v8f c_rect = {0.f, 0.f, 0.f, 0.f, 0.f, 0.f, 0.f, 0.f};

    for (int step = 0; step < WMMA_STEPS; ++step) {
        const int idx0 = wbase + step * 64 + lane * 2;   // coalesced: wave reads 256B
        if (idx0 + 512 < HWc)
            __builtin_prefetch(plane + idx0 + 512, 0, 0);
        v2f a = *(const v2f*)(plane + idx0);
        const int i0 = idx0 >> 7, j0 = idx0 & (Wc - 1);
        const int i1 = (idx0 + 1) >> 7, j1 = (idx0 + 1) & (Wc - 1);
        v2f am;
        am.x = (i0 >= ilo && i0 <= ihi && j0 >= jlo && j0 <= jhi) ? a.x : 0.0f;
        am.y = (i1 >= ilo && i1 <= ihi && j1 >= jlo && j1 <= jhi) ? a.y : 0.0f;
        c_tot  = __builtin_amdgcn_wmma_f32_16x16x4_f32(false, a,  false, bones,
                                                       (short)0, c_tot,  false, false);
        c_rect = __builtin_amdgcn_wmma_f32_16x16x4_f32(false, am, false, bones,
                                                       (short)0, c_rect, false, false);
    }

    // Per-lane fold of the 8 accumulator rows, then wave32 shuffle reduce.
    float tsum = c_tot[0] + c_tot[1] + c_tot[2] + c_tot[3] +
                 c_tot[4] + c_tot[5] + c_tot[6] + c_tot[7];
    float rsum = c_rect[0] + c_rect[1] + c_rect[2] + c_rect[3] +
                 c_rect[4] + c_rect[5] + c_rect[6] + c_rect[7];
    for (int off = 16; off >= 1; off >>= 1) {
        tsum += __shfl_down(tsum, off, 32);
        rsum += __shfl_down(rsum, off, 32);
    }
    if (lane == 0) {
        wtot[wave]  = tsum * 0.0625f;     // /16: each of 16 D-columns repeats the sum
        wrect[wave] = rsum * 0.0625f;
    }
    __syncthreads();
    if (threadIdx.x == 0) {
        float tot_s = 0.0f, rect_s = 0.0f;
        for (int w = 0; w < WAVES_PER_BLK; ++w) { tot_s += wtot[w]; rect_s += wrect[w]; }
        float in_sum  = (float)((ihi - ilo + 1) * (jhi - jlo + 1));
        float out_sum = (float)HWc - in_sum;
        float cam_in  = rect_s / fmaxf(in_sum, 1.0f);
        float cam_out = (tot_s - rect_s) / fmaxf(out_sum, 1.0f);
        float term = ((in_sum  > 0.0f) ? (1.0f - cam_in) : 0.0f) +
                     ((out_sum > 0.0f) ? cam_out : 0.0f);
        terms[pair] = term;
    }
}

// ---------------------------------------------------------------------------
// K6: final fixed-order reduction + scalar outputs
// ---------------------------------------------------------------------------
__global__ void k_finalize(const float* __restrict__ p_l1,
                           const float* __restrict__ p_gi,
                           const float* __restrict__ p_cf,
                           const int* __restrict__ p_np,
                           const float* __restrict__ terms,
                           float* __restrict__ out) {
    __shared__ float shf[TPB];
    __shared__ int shi[TPB];
    const int t = threadIdx.x;
    float a = 0.0f, g = 0.0f, cf = 0.0f;
    int np = 0;
    for (int i = t; i < NB_LOSS; i += TPB) {
        a += p_l1[i]; g += p_gi[i]; cf += p_cf[i]; np += p_np[i];
    }
    float camt = terms[t];   // exactly 256 entries

    float a_t   = blk_sum_f(a, shf);
    float g_t   = blk_sum_f(g, shf);
    float c_t   = blk_sum_f(cf, shf);
    float cam_t = blk_sum_f(camt, shf);
    int   np_t  = blk_sum_i(np, shi);

    if (t == 0) {
        float denom = fmaxf((float)np_t, 1.0f);
        float loss_l1   = a_t / denom;
        float loss_giou = g_t / denom;
        float loss_conf = c_t / (float)NCELLS;
        float loss_cam  = cam_t / (float)NPAIR;
        out[0] = 1.0f * loss_l1 + 2.0f * loss_giou + 1.0f * loss_conf + 0.5f * loss_cam;
        out[1] = loss_l1;
        out[2] = loss_giou;
        out[3] = loss_conf;
        out[4] = loss_cam;
    }
}

// ---------------------------------------------------------------------------
// Host launcher
// ---------------------------------------------------------------------------
extern "C" void kernel_launch(void* const* d_in, const int* in_sizes, int n_in,
                              void* d_out, int out_size, void* d_ws, size_t ws_size,
                              hipStream_t stream) {
    (void)in_sizes; (void)n_in; (void)out_size; (void)ws_size;
    const float* pred = (const float*)d_in[0];   // (B,C,H,W,4)
    const float* conf = (const float*)d_in[1];   // (B,C,H,W)
    const float* cam  = (const float*)d_in[2];   // (B,C,H,W)
    const float* gtb  = (const float*)d_in[3];   // (B,K,4)
    const int*   glab = (const int*)d_in[4];     // (B,K)
    float* out = (float*)d_out;                  // 5 floats

    // Workspace layout (~25.6 MB total)
    char* ws = (char*)d_ws;
    size_t off = 0;
    int* matched = (int*)(ws + off);   off += (size_t)NCELLS * sizeof(int);
    float* gmax  = (float*)(ws + off); off += NPAIR * sizeof(float);
    int*   amax  = (int*)(ws + off);   off += NPAIR * sizeof(int);
    float* p_l1  = (float*)(ws + off); off += NB_LOSS * sizeof(float);
    float* p_gi  = (float*)(ws + off); off += NB_LOSS * sizeof(float);
    float* p_cf  = (float*)(ws + off); off += NB_LOSS * sizeof(float);
    int*   p_np  = (int*)(ws + off);   off += NB_LOSS * sizeof(int);
    float* terms = (float*)(ws + off); off += NPAIR * sizeof(float);

    k_init    <<<NB_LOSS, TPB, 0, stream>>>(matched);
    k_match   <<<NPAIR,   TPB, 0, stream>>>(pred, gtb, glab, gmax, amax);
    k_scatter <<<1,       TPB, 0, stream>>>(gmax, amax, glab, matched);
    k_loss    <<<NB_LOSS, TPB, 0, stream>>>((const float4*)pred, conf,
                                            (const float4*)gtb, matched,
                                            p_l1, p_gi, p_cf, p_np);
    k_cam     <<<NPAIR,   TPB, 0, stream>>>(cam, gtb, glab, terms);
    k_finalize<<<1,       TPB, 0, stream>>>(p_l1, p_gi, p_cf, p_np, terms, out);
}